// ModelVLLM_70471823392998
// MI455X (gfx1250) — compile-verified
//
#include <hip/hip_runtime.h>

// reshape_and_cache_flash for MI455X (gfx1250): pure bandwidth kernel.
// Traffic: 256MB cache read + 256MB out write + 128MB kv read + ~108MB
// scatter write ~= 0.75GB -> ~32us floor at 23.3 TB/s. Zero FLOPs -> WMMA
// not applicable; the CDNA5 path that matters is the async global<->LDS
// DMA engine (ASYNCcnt). MLP: need ~18MB in flight (23.3TB/s x ~800ns), so
// each wave keeps 8 x 512B async B128 transfers outstanding per phase
// instead of one serialized round-trip.

typedef __attribute__((ext_vector_type(4))) float v4f;
typedef __attribute__((ext_vector_type(4))) int v4i;

#define AS1 __attribute__((address_space(1)))
#define AS3 __attribute__((address_space(3)))

#define UNROLL 8
#define CPY_THREADS 256

static __device__ __forceinline__ void wait_async0() {
  __builtin_amdgcn_s_wait_asynccnt(0);
}

// ---------------------------------------------------------------------------
// Bulk copy: both caches -> d_out in ONE launch. Blocks [0, bpt) copy K,
// [bpt, 2*bpt) copy V. Each thread batches UNROLL async global->LDS loads,
// waits once, then UNROLL async LDS->global stores (deep ASYNCcnt pipeline).
// ---------------------------------------------------------------------------
__global__ void cache_copy_kernel(const v4f* __restrict__ srcK,
                                  v4f* __restrict__ dstK,
                                  const v4f* __restrict__ srcV,
                                  v4f* __restrict__ dstV,
                                  unsigned long long n4,
                                  unsigned int blocks_per_tensor) {
  const bool isV = blockIdx.x >= blocks_per_tensor;
  const v4f* __restrict__ src = isV ? srcV : srcK;
  v4f* __restrict__ dst = isV ? dstV : dstK;
  const unsigned int b = isV ? (blockIdx.x - blocks_per_tensor) : blockIdx.x;

  unsigned long long idx = (unsigned long long)b * CPY_THREADS + threadIdx.x;
  const unsigned long long stride =
      (unsigned long long)blocks_per_tensor * CPY_THREADS;

  __shared__ __align__(16) unsigned char smem[CPY_THREADS * 16 * UNROLL];
  AS3 unsigned char* lds_base =
      (AS3 unsigned char*)smem + (unsigned)threadIdx.x * 16u;

  // Main unrolled pipeline: UNROLL outstanding loads, then UNROLL stores.
  while (idx + (unsigned long long)(UNROLL - 1) * stride < n4) {
#pragma unroll
    for (int u = 0; u < UNROLL; ++u) {
      __builtin_amdgcn_global_load_async_to_lds_b128(
          (AS1 v4i*)(src + idx + (unsigned long long)u * stride),
          (AS3 v4i*)(lds_base + u * (CPY_THREADS * 16)), 0, 0);
    }
    wait_async0();  // all UNROLL loads landed (loads complete in order)
#pragma unroll
    for (int u = 0; u < UNROLL; ++u) {
      __builtin_amdgcn_global_store_async_from_lds_b128(
          (AS1 v4i*)(dst + idx + (unsigned long long)u * stride),
          (AS3 v4i*)(lds_base + u * (CPY_THREADS * 16)), 0, 0);
    }
    wait_async0();  // LDS slots reusable; stores drain while others load
    idx += (unsigned long long)UNROLL * stride;
  }
  // Tail (not taken for the reference shapes: 8 strides exactly).
  for (; idx < n4; idx += stride) {
    __builtin_amdgcn_global_load_async_to_lds_b128(
        (AS1 v4i*)(src + idx), (AS3 v4i*)lds_base, 0, 0);
    wait_async0();
    __builtin_amdgcn_global_store_async_from_lds_b128(
        (AS1 v4i*)(dst + idx), (AS3 v4i*)lds_base, 0, 0);
    wait_async0();
  }
  wait_async0();
}

// ---------------------------------------------------------------------------
// Deterministic duplicate resolution: winner[slot] = max token index.
// ---------------------------------------------------------------------------
__global__ void winners_init_kernel(int* __restrict__ w, int n) {
  int i = blockIdx.x * blockDim.x + threadIdx.x;
  if (i < n) w[i] = -1;
}

__global__ void winners_mark_kernel(const int* __restrict__ slots,
                                    int* __restrict__ w,
                                    int ntok, int total_slots) {
  int t = blockIdx.x * blockDim.x + threadIdx.x;
  if (t >= ntok) return;
  int s = slots[t];
  if (s >= 0 && s < total_slots) atomicMax(&w[s], t);
}

// ---------------------------------------------------------------------------
// Scatter: one block per token; each token is f4_per_token contiguous float4s
// in both source (key/value) and destination (caches). One float4 per thread
// per tensor, non-temporal stores (write-once streaming output).
// ---------------------------------------------------------------------------
__global__ void scatter_kv_kernel(const v4f* __restrict__ key,
                                  const v4f* __restrict__ value,
                                  const int* __restrict__ slots,
                                  const int* __restrict__ w,
                                  v4f* __restrict__ out_k,
                                  v4f* __restrict__ out_v,
                                  int total_slots, int f4_per_token) {
  const int t = blockIdx.x;  // token
  const int s = slots[t];
  if (s < 0 || s >= total_slots) return;  // padded token: drop
  if (w[s] != t) return;                  // deterministic winner only
  const int e = threadIdx.x;              // float4 index within token
  if (e >= f4_per_token) return;
  const long long src = (long long)t * f4_per_token + e;
  const long long dst = (long long)s * f4_per_token + e;
  v4f kv = __builtin_nontemporal_load(key + src);
  v4f vv = __builtin_nontemporal_load(value + src);
  __builtin_nontemporal_store(kv, out_k + dst);
  __builtin_nontemporal_store(vv, out_v + dst);
}

extern "C" void kernel_launch(void* const* d_in, const int* in_sizes, int n_in,
                              void* d_out, int out_size, void* d_ws,
                              size_t ws_size, hipStream_t stream) {
  const float* key = (const float*)d_in[0];        // (T, H, D) f32
  const float* value = (const float*)d_in[1];      // (T, H, D) f32
  const float* key_cache = (const float*)d_in[2];  // (NB, BS, H, D) f32
  const float* value_cache = (const float*)d_in[3];
  const int* slot_mapping = (const int*)d_in[4];   // harness: integer -> int32
  // d_in[5]/d_in[6] = k_scale/v_scale: unused ('auto' dtype, no fp8 quant)

  const int num_tokens = in_sizes[4];
  const long long cache_elems = (long long)in_sizes[2];        // NB*BS*H*D
  const int f_per_token = (int)((long long)in_sizes[0] / num_tokens);  // H*D
  const int f4_per_token = f_per_token / 4;                    // 256
  const int total_slots = (int)(cache_elems / f_per_token);    // NB*BS
  const unsigned long long n4 = (unsigned long long)(cache_elems / 4);

  float* out_k = (float*)d_out;
  float* out_v = out_k + cache_elems;
  int* winners = (int*)d_ws;  // total_slots ints (128 KB)

  const dim3 blk(CPY_THREADS);
  const unsigned int blocks_per_tensor = 4096;  // 1M lanes -> 8 B128/lane

  // Phase 1: stream both caches into the output (one launch, K+V split).
  cache_copy_kernel<<<2 * blocks_per_tensor, blk, 0, stream>>>(
      (const v4f*)key_cache, (v4f*)out_k, (const v4f*)value_cache,
      (v4f*)out_v, n4, blocks_per_tensor);

  // Phase 2: deterministic winner per slot (last == max token index wins).
  winners_init_kernel<<<(total_slots + 255) / 256, blk, 0, stream>>>(
      winners, total_slots);
  winners_mark_kernel<<<(num_tokens + 255) / 256, blk, 0, stream>>>(
      slot_mapping, winners, num_tokens, total_slots);

  // Phase 3: scatter tokens into the copied caches.
  scatter_kv_kernel<<<num_tokens, blk, 0, stream>>>(
      (const v4f*)key, (const v4f*)value, slot_mapping, winners,
      (v4f*)out_k, (v4f*)out_v, total_slots, f4_per_token);
}